// CausalSelfAttention_86638080295181
// MI455X (gfx1250) — compile-verified
//
#include <hip/hip_runtime.h>

typedef __attribute__((ext_vector_type(2))) float v2f;
typedef __attribute__((ext_vector_type(8))) float v8f;

__device__ __forceinline__ v8f wmma_f32(v2f a, v2f b, v8f c) {
    // D = A(16x4 f32) * B(4x16 f32) + C(16x16 f32)
    return __builtin_amdgcn_wmma_f32_16x16x4_f32(
        /*neg_a=*/false, a, /*neg_b=*/false, b,
        /*c_mod=*/(short)0, c, /*reuse_a=*/false, /*reuse_b=*/false);
}

// ---- CDNA5 async global->LDS copies (ASYNCcnt-tracked, no VGPR round-trip) ----
__device__ __forceinline__ void async_b128(void* lds, const void* g) {
    // generic pointer to __shared__ = {aperture_hi, lds_offset}; low 32 bits
    // are the wave-relative LDS byte offset the instruction expects.
    uint32_t l = (uint32_t)(uintptr_t)lds;
    asm volatile("global_load_async_to_lds_b128 %0, %1, off"
                 :: "v"(l), "v"(g) : "memory");
}
__device__ __forceinline__ void async_b32(void* lds, const void* g) {
    uint32_t l = (uint32_t)(uintptr_t)lds;
    asm volatile("global_load_async_to_lds_b32 %0, %1, off"
                 :: "v"(l), "v"(g) : "memory");
}
__device__ __forceinline__ void wait_async_all() {
    asm volatile("s_wait_asynccnt 0x0" ::: "memory");
}
__device__ __forceinline__ void wait_async_le10() {   // one 10-op group in flight
    asm volatile("s_wait_asynccnt 0xa" ::: "memory");
}

// ---------------------------------------------------------------------------
// Tiled fp32 WMMA GEMM: C[M,N] = A[M,K] @ B[K,N]
// 256 threads (8 waves). Block tile 128x128, wave tile 64x32 (4x2 wmma tiles).
// K staged through double-buffered LDS in chunks of 16 (4 wmma k-steps) using
// CDNA5 async-to-LDS copies; async b32 performs the B transpose in flight.
// M,N,K assumed multiples of 128/128/16 (true for this problem).
// ---------------------------------------------------------------------------
#define KT 16
#define LP 20  // LDS pitch (floats): conflict-free, keeps float2/float4 alignment

__global__ __launch_bounds__(256) void gemm_f32_wmma(
    const float* __restrict__ A, const float* __restrict__ B,
    float* __restrict__ C, int M, int N, int K)
{
    __shared__ float As[2][128][LP];  // As[buf][row][k]
    __shared__ float Bs[2][128][LP];  // Bs[buf][n][k]  (B tile stored transposed)

    const int tid  = threadIdx.x;
    const int lane = tid & 31;
    const int wave = tid >> 5;
    const int l16  = lane & 15;
    const int hl   = lane >> 4;          // half-wave: selects K pair
    const int wm   = wave & 1;           // 2 M-groups of 64
    const int wn   = wave >> 1;          // 4 N-groups of 32

    const int bm = blockIdx.x * 128;
    const int bn = blockIdx.y * 128;

    v8f acc[4][2];
#pragma unroll
    for (int mt = 0; mt < 4; ++mt)
#pragma unroll
        for (int nt = 0; nt < 2; ++nt) acc[mt][nt] = (v8f){};

    // async staging assignments:
    //   A tile 128x16: thread -> (row, two 4-float chunks), 2 x b128
    //   B tile 16x128: thread -> (col n, 8 k's), 8 x b32, transposed into Bs[n][k]
    const int arow = tid >> 1;           // 0..127
    const int akf  = (tid & 1) * 8;      // first float of this thread's 8 (0 or 8)
    const int bcol = tid & 127;          // 0..127
    const int bkb  = (tid >> 7) * 8;     // 0 or 8

    const float* Ap = A + (size_t)(bm + arow) * K + akf;
    const float* Bp = B + (size_t)bn + bcol;

    auto issue = [&](int chunk, int buf) {
        const int k0 = chunk * KT;
        async_b128(&As[buf][arow][akf],     Ap + k0);
        async_b128(&As[buf][arow][akf + 4], Ap + k0 + 4);
        const float* bp = Bp + (size_t)(k0 + bkb) * N;
#pragma unroll
        for (int u = 0; u < 8; ++u)
            async_b32(&Bs[buf][bcol][bkb + u], bp + (size_t)u * N);
    };

    const int nchunks = K / KT;
    issue(0, 0);

    for (int chunk = 0; chunk < nchunks; ++chunk) {
        const int buf = chunk & 1;
        const bool pre = (chunk + 1 < nchunks);
        if (pre) issue(chunk + 1, buf ^ 1);       // overlaps with compute below
        if (pre) wait_async_le10(); else wait_async_all();  // group(chunk) done
        __syncthreads();                          // visible to all waves

#pragma unroll
        for (int kk = 0; kk < 4; ++kk) {
            const int kf = kk * 4 + hl * 2;       // K offset of this half's pair
            v2f af[4], bf[2];
#pragma unroll
            for (int mt = 0; mt < 4; ++mt)
                af[mt] = *(const v2f*)&As[buf][wm * 64 + mt * 16 + l16][kf];
#pragma unroll
            for (int nt = 0; nt < 2; ++nt)
                bf[nt] = *(const v2f*)&Bs[buf][wn * 32 + nt * 16 + l16][kf];
#pragma unroll
            for (int mt = 0; mt < 4; ++mt)
#pragma unroll
                for (int nt = 0; nt < 2; ++nt)
                    acc[mt][nt] = wmma_f32(af[mt], bf[nt], acc[mt][nt]);
        }
        __syncthreads();   // all waves done reading buf before it is refilled
    }

    // epilogue: C/D layout -> reg r, lane: row = r + 8*hl, col = l16
#pragma unroll
    for (int mt = 0; mt < 4; ++mt)
#pragma unroll
        for (int nt = 0; nt < 2; ++nt) {
            const int col = bn + wn * 32 + nt * 16 + l16;
#pragma unroll
            for (int r = 0; r < 8; ++r) {
                const int row = bm + wm * 64 + mt * 16 + r + 8 * hl;
                C[(size_t)row * N + col] = acc[mt][nt][r];
            }
        }
}

// ---------------------------------------------------------------------------
// Banded causal flash-attention. One wave handles one 16-query tile of one
// (batch, head). Band: key j visible to query i iff i-256 <= j <= i, so a
// 16-query tile touches exactly 17 aligned 16-key tiles.
// ---------------------------------------------------------------------------
#define AH   16   // heads
#define AD   64   // head dim
#define AWIN 256

__global__ __launch_bounds__(128) void banded_attn(
    const float* __restrict__ qkv, float* __restrict__ y, int Bn, int T)
{
    const int C  = AH * AD;      // 1024
    const int C3 = 3 * C;        // 3072

    __shared__ float pbuf[4][16][LP];   // per-wave P-transpose staging

    const int lane = threadIdx.x & 31;
    const int wave = threadIdx.x >> 5;
    const int l16  = lane & 15;
    const int hl   = lane >> 4;

    const int tilesPerSeq = T / 16;
    const int tile = blockIdx.x * 4 + wave;
    const int b  = tile / (AH * tilesPerSeq);
    const int h  = (tile / tilesPerSeq) % AH;
    const int i0 = (tile % tilesPerSeq) * 16;

    const float* base = qkv + (size_t)b * T * C3;

    // Q fragments for all 16 k-steps (covers D=64), kept in registers
    const float* qrow = base + (size_t)(i0 + l16) * C3 + h * AD;
    v2f qf[16];
#pragma unroll
    for (int s = 0; s < 16; ++s)
        qf[s] = *(const v2f*)(qrow + 4 * s + 2 * hl);

    float mrun[8], lrun[8];
#pragma unroll
    for (int r = 0; r < 8; ++r) { mrun[r] = -3.0e38f; lrun[r] = 0.0f; }
    v8f o[4];
#pragma unroll
    for (int nt = 0; nt < 4; ++nt) o[nt] = (v8f){};

    const float scale = 0.125f;  // 1/sqrt(64)
    const int jlo = i0 - AWIN;

    for (int t = 0; t < 17; ++t) {
        const int j0 = jlo + 16 * t;
        if (j0 < 0) continue;            // j0 is a multiple of 16: whole tile OOB

        // ---- S = Q * K^T (16 wmma, K=64) ----
        v8f s = (v8f){};
        const float* krow = base + C + (size_t)(j0 + l16) * C3 + h * AD;
#pragma unroll
        for (int ss = 0; ss < 16; ++ss) {
            v2f kf = *(const v2f*)(krow + 4 * ss + 2 * hl);
            s = wmma_f32(qf[ss], kf, s);
        }

        // ---- band mask + online softmax (row stats via 16-lane shuffles) ----
#pragma unroll
        for (int r = 0; r < 8; ++r) {
            const int i = i0 + r + 8 * hl;     // query row of this element
            const int j = j0 + l16;            // key col of this element
            float sv = ((j <= i) && (i - j <= AWIN)) ? s[r] * scale : -3.0e38f;
            float mx = sv;
            mx = fmaxf(mx, __shfl_xor(mx, 1));
            mx = fmaxf(mx, __shfl_xor(mx, 2));
            mx = fmaxf(mx, __shfl_xor(mx, 4));
            mx = fmaxf(mx, __shfl_xor(mx, 8));
            const float mnew = fmaxf(mrun[r], mx);
            const float corr = __expf(mrun[r] - mnew);
            const float e = __expf(sv - mnew);
            float es = e;
            es += __shfl_xor(es, 1);
            es += __shfl_xor(es, 2);
            es += __shfl_xor(es, 4);
            es += __shfl_xor(es, 8);
            lrun[r] = lrun[r] * corr + es;
            mrun[r] = mnew;
            o[0][r] *= corr; o[1][r] *= corr; o[2][r] *= corr; o[3][r] *= corr;
            s[r] = e;                          // probability
        }

        // ---- transpose P (C/D layout) -> A-operand layout via LDS ----
#pragma unroll
        for (int r = 0; r < 8; ++r)
            pbuf[wave][r + 8 * hl][l16] = s[r];
        // same-wave DS ops are ordered; no barrier needed (private buffer)

        // ---- O += P * V (16 wmma, K=16, N=64) ----
        const float* vbase = base + 2 * C + h * AD;
#pragma unroll
        for (int kk = 0; kk < 4; ++kk) {
            v2f af = *(const v2f*)&pbuf[wave][l16][4 * kk + 2 * hl];
            const float* v0 = vbase + (size_t)(j0 + 4 * kk + 2 * hl) * C3;
#pragma unroll
            for (int nt = 0; nt < 4; ++nt) {
                const int d = nt * 16 + l16;
                v2f bf;
                bf.x = v0[d];
                bf.y = v0[C3 + d];
                o[nt] = wmma_f32(af, bf, o[nt]);
            }
        }
    }

    // ---- normalize and write this head's slice of y [B*T, C] ----
#pragma unroll
    for (int nt = 0; nt < 4; ++nt)
#pragma unroll
        for (int r = 0; r < 8; ++r) {
            const int m = r + 8 * hl;
            const int d = nt * 16 + l16;
            y[(size_t)(b * T + i0 + m) * C + h * AD + d] = o[nt][r] / lrun[r];
        }
}

// ---------------------------------------------------------------------------
extern "C" void kernel_launch(void* const* d_in, const int* in_sizes, int n_in,
                              void* d_out, int out_size, void* d_ws, size_t ws_size,
                              hipStream_t stream) {
    const float* x      = (const float*)d_in[0];  // [B,T,C]
    const float* W_attn = (const float*)d_in[1];  // [C,3C]
    const float* W_proj = (const float*)d_in[2];  // [C,C]
    float* out = (float*)d_out;

    const int Bn = 2, T = 2048, C = 1024;         // matches reference setup
    const int M = Bn * T;                         // 4096

    float* qkv = (float*)d_ws;                    // M * 3C floats (48 MB)
    float* yw  = qkv + (size_t)M * 3 * C;         // M * C floats  (16 MB)

    dim3 blk(256);
    // 1) qkv = x @ W_attn
    gemm_f32_wmma<<<dim3(M / 128, (3 * C) / 128), blk, 0, stream>>>(
        x, W_attn, qkv, M, 3 * C, C);
    // 2) banded attention -> yw
    const int nTiles = Bn * AH * (T / 16);        // 4096 query tiles
    banded_attn<<<dim3(nTiles / 4), dim3(128), 0, stream>>>(qkv, yw, Bn, T);
    // 3) out = yw @ W_proj
    gemm_f32_wmma<<<dim3(M / 128, C / 128), blk, 0, stream>>>(
        yw, W_proj, out, M, C, C);
}